// NoopSegmenter_35012573397109
// MI455X (gfx1250) — compile-verified
//
#include <hip/hip_runtime.h>

#define BATCH 64
#define LFR   500      // frames
#define NB    501      // boundary slots
#define DDIM  1024
#define SMAX  50

typedef __attribute__((ext_vector_type(2))) float v2f;
typedef __attribute__((ext_vector_type(8))) float v8f;

// ---------------------------------------------------------------------------
// Kernel 1: per-batch segmentation metadata.
//   segid[b][l] : segment index of frame l
//   w[b][l]     : 1/len if frame's segment is closed and seg<50, else 0
//   mask_sum[b][s] = 1.0 for closed segments s<50, else 0
//   bnd_out = pass-through copy of in_boundary
// ---------------------------------------------------------------------------
__global__ void seg_prep(const float* __restrict__ in_boundary,
                         int* __restrict__ segid, float* __restrict__ w,
                         float* __restrict__ mask_sum, float* __restrict__ bnd_out)
{
    int b = blockIdx.x;
    const float* ib = in_boundary + b * NB;

    for (int t = threadIdx.x; t < NB; t += blockDim.x)
        bnd_out[b * NB + t] = ib[t];
    for (int s = threadIdx.x; s < SMAX; s += blockDim.x)
        mask_sum[b * SMAX + s] = 0.0f;
    __syncthreads();

    if (threadIdx.x == 0) {
        int*   sg = segid + b * LFR;
        float* wb = w     + b * LFR;
        int seg = 0, start = 0;                    // boundary forced at t=0
        for (int t = 1; t <= LFR; ++t) {
            if (ib[t] != 0.0f) {                   // segment [start, t) closes
                float wv = (seg < SMAX) ? (1.0f / (float)(t - start)) : 0.0f;
                for (int l = start; l < t; ++l) { sg[l] = seg; wb[l] = wv; }
                if (seg < SMAX) mask_sum[b * SMAX + seg] = 1.0f;
                ++seg; start = t;
            }
        }
        for (int l = start; l < LFR; ++l) { sg[l] = seg; wb[l] = 0.0f; }
    }
}

// ---------------------------------------------------------------------------
// Kernel 2: out[b] = mask[b](50x500) @ x[b](500x1024) via V_WMMA_F32_16X16X4_F32.
// One wave per (batch, 16-col tile); 4 accumulators cover padded M=64 so each
// x element is read from global exactly once. segid/w staged in LDS (4KB).
//
// fp32 WMMA layouts (ISA 7.12.2):
//   A 16x4 : M = lane&15,  K = vgpr + 2*(lane>=16)
//   B 4x16 : N = lane&15,  K = vgpr + 2*(lane>=16)
//   C 16x16: N = lane&15,  M = vgpr + 8*(lane>=16)
// ---------------------------------------------------------------------------
__global__ void seg_gemm_wmma(const float* __restrict__ x,
                              const int* __restrict__ segid,
                              const float* __restrict__ w,
                              float* __restrict__ out)
{
    __shared__ int   s_sg[LFR];
    __shared__ float s_w [LFR];

    const int b    = blockIdx.y;
    const int wave = threadIdx.x >> 5;
    const int lane = threadIdx.x & 31;
    const int n_tile = blockIdx.x * (blockDim.x >> 5) + wave;   // 0..63
    const int khi  = lane >> 4;          // K-group: lanes 16-31 hold K+2
    const int lo   = lane & 15;
    const int n    = n_tile * 16 + lo;   // output column

    // stage per-batch segid/weight into LDS once
    {
        const int*   sg = segid + b * LFR;
        const float* wb = w     + b * LFR;
        for (int i = threadIdx.x; i < LFR; i += blockDim.x) {
            s_sg[i] = sg[i];
            s_w [i] = wb[i];
        }
    }
    __syncthreads();

    const float* xb = x + (size_t)b * LFR * DDIM;

    v8f c0 = {}, c1 = {}, c2 = {}, c3 = {};
    const int m0 = lo, m1 = 16 + lo, m2 = 32 + lo, m3 = 48 + lo;

    for (int kb = 0; kb < LFR; kb += 4) {
        const int k0 = kb + 2 * khi;
        const int k1 = k0 + 1;

        const int   sA = s_sg[k0], sB = s_sg[k1];
        const float wA = s_w[k0],  wB = s_w[k1];

        v2f Bv;
        Bv.x = xb[(size_t)k0 * DDIM + n];
        Bv.y = xb[(size_t)k1 * DDIM + n];

        v2f A;
        A.x = (sA == m0) ? wA : 0.0f;  A.y = (sB == m0) ? wB : 0.0f;
        c0 = __builtin_amdgcn_wmma_f32_16x16x4_f32(false, A, false, Bv, (short)0, c0, false, false);
        A.x = (sA == m1) ? wA : 0.0f;  A.y = (sB == m1) ? wB : 0.0f;
        c1 = __builtin_amdgcn_wmma_f32_16x16x4_f32(false, A, false, Bv, (short)0, c1, false, false);
        A.x = (sA == m2) ? wA : 0.0f;  A.y = (sB == m2) ? wB : 0.0f;
        c2 = __builtin_amdgcn_wmma_f32_16x16x4_f32(false, A, false, Bv, (short)0, c2, false, false);
        A.x = (sA == m3) ? wA : 0.0f;  A.y = (sB == m3) ? wB : 0.0f;
        c3 = __builtin_amdgcn_wmma_f32_16x16x4_f32(false, A, false, Bv, (short)0, c3, false, false);
    }

    // store: row M = mtile*16 + i + 8*khi, col n; only M < 50 is real output
    float* ob = out + (size_t)b * SMAX * DDIM;
#pragma unroll
    for (int i = 0; i < 8; ++i) {
        const int r = i + 8 * khi;                 // 0..15 within a 16-row tile
        ob[(size_t)(r)      * DDIM + n] = c0[i];   // rows  0..15
        ob[(size_t)(16 + r) * DDIM + n] = c1[i];   // rows 16..31
        ob[(size_t)(32 + r) * DDIM + n] = c2[i];   // rows 32..47
        if (48 + r < SMAX)                          // rows 48..49 only
            ob[(size_t)(48 + r) * DDIM + n] = c3[i];
    }
}

// ---------------------------------------------------------------------------
extern "C" void kernel_launch(void* const* d_in, const int* in_sizes, int n_in,
                              void* d_out, int out_size, void* d_ws, size_t ws_size,
                              hipStream_t stream)
{
    (void)in_sizes; (void)n_in; (void)out_size; (void)ws_size;

    const float* x           = (const float*)d_in[0];   // [64,500,1024] f32
    const float* in_boundary = (const float*)d_in[1];   // [64,501] f32

    // d_out layout: out [64,50,1024] | mask_sum [64,50] | in_boundary copy [64,501]
    float* out      = (float*)d_out;
    float* mask_sum = out + (size_t)BATCH * SMAX * DDIM;
    float* bnd_out  = mask_sum + (size_t)BATCH * SMAX;

    // workspace: segid int[64*500] then w float[64*500] (256 KB total)
    int*   segid = (int*)d_ws;
    float* w     = (float*)((char*)d_ws + (size_t)BATCH * LFR * sizeof(int));

    seg_prep<<<BATCH, 64, 0, stream>>>(in_boundary, segid, w, mask_sum, bnd_out);
    seg_gemm_wmma<<<dim3(8, BATCH), 256, 0, stream>>>(x, segid, w, out);
}